// GroupConvTranspose3d_20005957664849
// MI455X (gfx1250) — compile-verified
//
#include <hip/hip_runtime.h>
#include <hip/hip_bf16.h>

typedef __attribute__((ext_vector_type(2))) float v2f;
typedef __attribute__((ext_vector_type(8))) float v8f;

// Problem:
// x:      [32(bg), 128(c), 4096(dhw)]   (bg = b*16+g flattened)
// kernel: [128(c), 1024(m)]  m = ((o*2+i)*2+j)*2+k
// out:    [32(bg), 128(o), 32, 32, 32]
// GEMM per bg: Y[m,n] = sum_c K[c,m] * X[c,n], then scatter-store.
#define X_BG_STRIDE   524288u   // 128*4096
#define OUT_BG_STRIDE 4194304u  // 128*32768

__device__ __forceinline__ void store_tile(float* __restrict__ out,
                                           const v8f& acc,
                                           unsigned bg, int o, int dh, int lw) {
    const int d = dh >> 4;
    const int h = dh & 15;
    float* po = out + (size_t)bg * OUT_BG_STRIDE + (size_t)o * 32768u
                    + d * 2048 + h * 64 + 2 * lw;
    // lane holds ijk = acc element index; (i,j) pair -> b64 store of k=0,1
    v2f p00; p00.x = acc[0]; p00.y = acc[1];   // i=0,j=0
    v2f p01; p01.x = acc[2]; p01.y = acc[3];   // i=0,j=1
    v2f p10; p10.x = acc[4]; p10.y = acc[5];   // i=1,j=0
    v2f p11; p11.x = acc[6]; p11.y = acc[7];   // i=1,j=1
    *(v2f*)(po)             = p00;
    *(v2f*)(po + 32)        = p01;
    *(v2f*)(po + 1024)      = p10;
    *(v2f*)(po + 1024 + 32) = p11;
}

__global__ __launch_bounds__(256) void
GroupConvTranspose3d_20005957664849_kernel(const float* __restrict__ x,
                                           const float* __restrict__ kw,
                                           float* __restrict__ out) {
    const int lane = threadIdx.x & 31;
    const unsigned wave = blockIdx.x * 8u + (threadIdx.x >> 5);

    // 32 bg * 16 M-blocks(64 rows) * 64 N-blocks(64 cols) = 32768 waves
    const unsigned nb = wave & 63u;
    const unsigned mb = (wave >> 6) & 15u;
    const unsigned bg = wave >> 10;

    const int half = lane >> 4;              // 0: K rows c0,c0+1 ; 1: c0+2,c0+3
    const int lw   = lane & 15;

    const int nbase = nb * 64;
    const int mbase = mb * 64;
    const int dh0   = nbase >> 4;            // 4 consecutive (d*16+h) rows

    // B source: X[bg][c][dh*16 + w], lane supplies column w=lw
    const float* xp = x + (size_t)bg * X_BG_STRIDE
                        + (size_t)(2 * half) * 4096 + dh0 * 16 + lw;
    // A source: kernel[c][mbase + l]  (A = K^T, lane supplies row m)
    const float* ap = kw + (size_t)(2 * half) * 1024 + mbase + lw;

    v8f acc[16];
#pragma unroll
    for (int t = 0; t < 16; ++t) acc[t] = (v8f){};

#pragma unroll 2
    for (int kt = 0; kt < 32; ++kt) {
        const float* xk = xp + kt * 4 * 4096;
        const float* ak = ap + kt * 4 * 1024;

        v2f a[4], b[4];
#pragma unroll
        for (int t = 0; t < 4; ++t) {
            // B fragment t: N-tile = (dh0+t) w-row. VGPR0=K row c0+2h, VGPR1=c0+1+2h
            b[t].x = xk[t * 16];
            b[t].y = xk[4096 + t * 16];
            // A fragment t: M-tile rows mbase+16t .. +15
            a[t].x = ak[t * 16];
            a[t].y = ak[1024 + t * 16];
        }
#pragma unroll
        for (int tm = 0; tm < 4; ++tm)
#pragma unroll
            for (int tn = 0; tn < 4; ++tn)
                acc[tm * 4 + tn] = __builtin_amdgcn_wmma_f32_16x16x4_f32(
                    false, a[tm], false, b[tn], (short)0, acc[tm * 4 + tn],
                    false, false);
    }

    // D layout: lane holds o = (tile o base) + half, ijk = element idx, col = lw
    const int obase = mbase >> 3;            // 8 consecutive o per M-block
#pragma unroll
    for (int tm = 0; tm < 4; ++tm)
#pragma unroll
        for (int tn = 0; tn < 4; ++tn)
            store_tile(out, acc[tm * 4 + tn], bg,
                       obase + 2 * tm + half, dh0 + tn, lw);
}

extern "C" void kernel_launch(void* const* d_in, const int* in_sizes, int n_in,
                              void* d_out, int out_size, void* d_ws, size_t ws_size,
                              hipStream_t stream) {
    const float* x  = (const float*)d_in[0];   // [2,16,128,16,16,16] f32
    const float* kw = (const float*)d_in[1];   // [128,128,2,2,2]     f32
    float* out = (float*)d_out;                // [2,16,128,32,32,32] f32

    // 32768 waves / 8 waves per 256-thread block = 4096 blocks
    dim3 grid(4096), block(256);
    hipLaunchKernelGGL(GroupConvTranspose3d_20005957664849_kernel,
                       grid, block, 0, stream, x, kw, out);
    (void)in_sizes; (void)n_in; (void)out_size; (void)d_ws; (void)ws_size;
}